// Block_5617817223712
// MI455X (gfx1250) — compile-verified
//
#include <hip/hip_runtime.h>
#include <math.h>

#define L_SEQ 4096
#define NBATCH 2
#define DIM 96
#define DI 192
#define HID 192

typedef __attribute__((ext_vector_type(2))) float v2f;
typedef __attribute__((ext_vector_type(8))) float v8f;

static __device__ __forceinline__ v8f wmma_f32_k4(v2f a, v2f b, v8f c) {
  // V_WMMA_F32_16X16X4_F32 : D = A(16x4,f32) x B(4x16,f32) + C(16x16,f32)
  return __builtin_amdgcn_wmma_f32_16x16x4_f32(false, a, false, b, (short)0, c, false, false);
}

static __device__ __forceinline__ float wave_sum32(float v) {
#pragma unroll
  for (int off = 16; off >= 1; off >>= 1) v += __shfl_xor(v, off, 32);
  return v;
}

static __device__ __forceinline__ float siluf(float x) {
  return x / (1.f + expf(-x));
}

// -------- LayerNorm over channels (channels-first, planar [B][96][L]) --------
// out1 = ln_cf(x, w1, b1, eps=1e-6); out2 (optional) = ln(out1, w2, b2, eps=1e-5)
__global__ void ln_kernel(const float* __restrict__ xin,
                          const float* __restrict__ w1, const float* __restrict__ b1,
                          const float* __restrict__ w2, const float* __restrict__ b2,
                          float* __restrict__ out1, float* __restrict__ out2) {
  int wid  = (blockIdx.x * blockDim.x + threadIdx.x) >> 5;  // one wave per position
  int lane = threadIdx.x & 31;
  int b = wid >> 12;
  int l = wid & (L_SEQ - 1);
  size_t base = ((size_t)b * DIM) * L_SEQ + l;
  const float* px = xin + base;
  float v0 = px[0];
  float v1 = px[(size_t)32 * L_SEQ];
  float v2 = px[(size_t)64 * L_SEQ];
  float mu = wave_sum32(v0 + v1 + v2) * (1.f / 96.f);
  float d0 = v0 - mu, d1 = v1 - mu, d2 = v2 - mu;
  float var = wave_sum32(d0 * d0 + d1 * d1 + d2 * d2) * (1.f / 96.f);
  float inv = rsqrtf(var + 1e-6f);
  float y0 = w1[lane]      * d0 * inv + b1[lane];
  float y1 = w1[lane + 32] * d1 * inv + b1[lane + 32];
  float y2 = w1[lane + 64] * d2 * inv + b1[lane + 64];
  float* p1 = out1 + base;
  p1[0] = y0; p1[(size_t)32 * L_SEQ] = y1; p1[(size_t)64 * L_SEQ] = y2;
  if (out2 != nullptr) {
    float mu2 = wave_sum32(y0 + y1 + y2) * (1.f / 96.f);
    float e0 = y0 - mu2, e1 = y1 - mu2, e2 = y2 - mu2;
    float var2 = wave_sum32(e0 * e0 + e1 * e1 + e2 * e2) * (1.f / 96.f);
    float inv2 = rsqrtf(var2 + 1e-5f);
    float* p2 = out2 + base;
    p2[0]                    = e0 * inv2 * w2[lane]      + b2[lane];
    p2[(size_t)32 * L_SEQ]   = e1 * inv2 * w2[lane + 32] + b2[lane + 32];
    p2[(size_t)64 * L_SEQ]   = e2 * inv2 * w2[lane + 64] + b2[lane + 64];
  }
}

// -------- Generic weight x planar-activation GEMM --------
// D[b][m][l] = sum_k A[m][k] * Bact[b][k][l] ; one wave = 16M x 64N strip.
__global__ void gemm_wact_kernel(const float* __restrict__ A,
                                 const float* __restrict__ Bact,
                                 float* __restrict__ D, int M, int K) {
  int gw   = (blockIdx.x * blockDim.x + threadIdx.x) >> 5;
  int lane = threadIdx.x & 31;
  int lh = lane >> 4, l16 = lane & 15;
  int mtiles = M >> 4;
  int per_b = mtiles * (L_SEQ / 64);
  int b = gw / per_b;
  int r = gw % per_b;
  int mt = r / (L_SEQ / 64);
  int ns = r % (L_SEQ / 64);
  int m0 = mt << 4, n0 = ns << 6;
  v8f acc[4] = {};
  const float* Ap = A + (size_t)(m0 + l16) * K + 2 * lh;
  const float* Bp = Bact + (size_t)b * K * L_SEQ + n0 + l16;
  for (int k = 0; k < K; k += 4) {
    v2f a; a.x = Ap[k]; a.y = Ap[k + 1];
    const float* bp = Bp + (size_t)(k + 2 * lh) * L_SEQ;
#pragma unroll
    for (int nb = 0; nb < 4; ++nb) {
      v2f bb; bb.x = bp[nb * 16]; bb.y = bp[L_SEQ + nb * 16];
      acc[nb] = wmma_f32_k4(a, bb, acc[nb]);
    }
  }
  float* Dp = D + (size_t)b * M * L_SEQ + (size_t)(m0 + 8 * lh) * L_SEQ + n0 + l16;
#pragma unroll
  for (int rr = 0; rr < 8; ++rr)
#pragma unroll
    for (int nb = 0; nb < 4; ++nb)
      Dp[(size_t)rr * L_SEQ + nb * 16] = acc[nb][rr];
}

// -------- Depthwise causal/anti-causal conv1d (K=4) + SiLU --------
__global__ void conv1d_silu_kernel(const float* __restrict__ xzT,
                                   const float* __restrict__ cw,
                                   const float* __restrict__ cb,
                                   float* __restrict__ xsT, int rev) {
  int idx = blockIdx.x * blockDim.x + threadIdx.x;
  int l = idx & (L_SEQ - 1);
  int d = (idx >> 12) % DI;
  int b = idx / (DI * L_SEQ);
  const float* p = xzT + ((size_t)b * 2 * DI + d) * L_SEQ;  // xc planes 0..191
  const float* w = cw + d * 4;
  float acc = cb[d];
  if (!rev) {
#pragma unroll
    for (int j = 0; j < 4; ++j) { int li = l - 3 + j; if (li >= 0) acc += w[j] * p[li]; }
  } else {
#pragma unroll
    for (int j = 0; j < 4; ++j) { int li = l + 3 - j; if (li < L_SEQ) acc += w[j] * p[li]; }
  }
  xsT[((size_t)b * DI + d) * L_SEQ + l] = siluf(acc);
}

// -------- x_dbl = xs @ Wx.T  (M=positions, N=38->48 padded, K=192) --------
__global__ void gemm_xdbl_kernel(const float* __restrict__ xsT,
                                 const float* __restrict__ Wx,
                                 float* __restrict__ xdbl) {
  const int K = DI;
  int gw   = (blockIdx.x * blockDim.x + threadIdx.x) >> 5;  // 0..511
  int lane = threadIdx.x & 31;
  int lh = lane >> 4, l16 = lane & 15;
  int m0 = gw << 4;
  int b = m0 >> 12;
  int l0 = m0 & (L_SEQ - 1);
  v8f acc[3] = {};
  const float* Ap = xsT + (size_t)b * DI * L_SEQ + l0 + l16;  // A[m][k] = xsT[b][k][l]
  for (int k = 0; k < K; k += 4) {
    int ka = k + 2 * lh;
    v2f a; a.x = Ap[(size_t)ka * L_SEQ]; a.y = Ap[(size_t)(ka + 1) * L_SEQ];
#pragma unroll
    for (int nb = 0; nb < 3; ++nb) {
      int n = nb * 16 + l16;
      v2f bb; bb.x = 0.f; bb.y = 0.f;
      if (n < 38) { bb.x = Wx[n * K + ka]; bb.y = Wx[n * K + ka + 1]; }
      acc[nb] = wmma_f32_k4(a, bb, acc[nb]);
    }
  }
  float* Db = xdbl + (size_t)b * L_SEQ * 38;
#pragma unroll
  for (int rr = 0; rr < 8; ++rr) {
    int lrow = l0 + rr + 8 * lh;
#pragma unroll
    for (int nb = 0; nb < 3; ++nb) {
      int n = nb * 16 + l16;
      if (n < 38) Db[(size_t)lrow * 38 + n] = acc[nb][rr];
    }
  }
}

// -------- Selective scan (both directions in one launch) --------
// lane s<16 per channel handles one SSM state; half-wave shfl reduction for h.C
__global__ void scan_kernel(const float* __restrict__ xsf, const float* __restrict__ xdf,
                            const float* __restrict__ xzf, const float* __restrict__ Wdtf,
                            const float* __restrict__ bdtf, const float* __restrict__ Alogf,
                            const float* __restrict__ Dpf,
                            const float* __restrict__ xsb, const float* __restrict__ xdb2,
                            const float* __restrict__ xzb, const float* __restrict__ Wdtb,
                            const float* __restrict__ bdtb, const float* __restrict__ Alogb,
                            const float* __restrict__ Dpb,
                            float* __restrict__ yf, float* __restrict__ yb) {
  int wid  = (blockIdx.x * blockDim.x + threadIdx.x) >> 5;  // 0..383
  int lane = threadIdx.x & 31;
  int lh = lane >> 4, s = lane & 15;
  int dir = wid / (NBATCH * 96);
  int r   = wid % (NBATCH * 96);
  int b = r / 96;
  int d = (r % 96) * 2 + lh;

  const float* xsT  = dir ? xsb : xsf;
  const float* xdbl = dir ? xdb2 : xdf;
  const float* xzT  = dir ? xzb : xzf;
  const float* Wdt  = dir ? Wdtb : Wdtf;
  const float* bdt  = dir ? bdtb : bdtf;
  const float* Alog = dir ? Alogb : Alogf;
  const float* Dp   = dir ? Dpb : Dpf;
  float* yT         = dir ? yb : yf;

  float Av = -expf(Alog[d * 16 + s]);
  float w0 = Wdt[d * 6 + 0], w1 = Wdt[d * 6 + 1], w2 = Wdt[d * 6 + 2];
  float w3 = Wdt[d * 6 + 3], w4 = Wdt[d * 6 + 4], w5 = Wdt[d * 6 + 5];
  float bd = bdt[d], Dv = Dp[d];
  const float* xsp  = xsT + ((size_t)b * DI + d) * L_SEQ;
  const float* zp   = xzT + ((size_t)b * 2 * DI + DI + d) * L_SEQ;
  const float* rowb = xdbl + (size_t)b * L_SEQ * 38;
  float* yp = yT + ((size_t)b * DI + d) * L_SEQ;

  float h = 0.f;
  int step = dir ? -1 : 1;
  int l = dir ? (L_SEQ - 1) : 0;
  for (int t = 0; t < L_SEQ; ++t, l += step) {
    const float* row = rowb + (size_t)l * 38;
    __builtin_prefetch(row + step * 38, 0, 1);
    float din = bd + w0 * row[0] + w1 * row[1] + w2 * row[2]
                   + w3 * row[3] + w4 * row[4] + w5 * row[5];
    float delta = (din > 20.f) ? din : log1pf(expf(din));
    float xv = xsp[l];
    float Bm = row[6 + s], Cm = row[22 + s];
    h = expf(delta * Av) * h + delta * Bm * xv;
    float c = h * Cm;
    c += __shfl_xor(c, 8, 16);
    c += __shfl_xor(c, 4, 16);
    c += __shfl_xor(c, 2, 16);
    c += __shfl_xor(c, 1, 16);
    if (s == 0) {
      float y  = c + xv * Dv;
      float zv = zp[l];
      yp[l] = y * siluf(zv);
    }
  }
}

// -------- Fused output GEMMs (M=96, K=192) with residual epilogue --------
// Dout[b][c][l] = xres[b][c][l] + gamma[c] * (sum_k A[c][k]*(B1+B2)[b][k][l] + xadd[b][c][l])
__global__ void gemm_fuse_kernel(const float* __restrict__ A,
                                 const float* __restrict__ B1,
                                 const float* __restrict__ B2,
                                 const float* __restrict__ xres,
                                 const float* __restrict__ xadd,
                                 const float* __restrict__ gamma,
                                 float* __restrict__ Dout, int K) {
  int gw   = (blockIdx.x * blockDim.x + threadIdx.x) >> 5;
  int lane = threadIdx.x & 31;
  int lh = lane >> 4, l16 = lane & 15;
  int per_b = 6 * (L_SEQ / 64);
  int b = gw / per_b;
  int r = gw % per_b;
  int mt = r / (L_SEQ / 64);
  int ns = r % (L_SEQ / 64);
  int m0 = mt << 4, n0 = ns << 6;
  v8f acc[4] = {};
  const float* Ap  = A + (size_t)(m0 + l16) * K + 2 * lh;
  const float* bp1 = B1 + (size_t)b * K * L_SEQ + n0 + l16;
  const float* bp2 = (B2 ? B2 : B1) + (size_t)b * K * L_SEQ + n0 + l16;
  float s2 = B2 ? 1.f : 0.f;
  for (int k = 0; k < K; k += 4) {
    v2f a; a.x = Ap[k]; a.y = Ap[k + 1];
    size_t ko = (size_t)(k + 2 * lh) * L_SEQ;
#pragma unroll
    for (int nb = 0; nb < 4; ++nb) {
      v2f bb;
      bb.x = bp1[ko + nb * 16]         + s2 * bp2[ko + nb * 16];
      bb.y = bp1[ko + L_SEQ + nb * 16] + s2 * bp2[ko + L_SEQ + nb * 16];
      acc[nb] = wmma_f32_k4(a, bb, acc[nb]);
    }
  }
  const float* xa = xadd ? xadd : xres;
  float sa = xadd ? 1.f : 0.f;
#pragma unroll
  for (int rr = 0; rr < 8; ++rr) {
    int c = m0 + 8 * lh + rr;
    float g = gamma[c];
    size_t rowo = ((size_t)b * DIM + c) * L_SEQ + n0 + l16;
#pragma unroll
    for (int nb = 0; nb < 4; ++nb) {
      size_t o = rowo + nb * 16;
      float v = acc[nb][rr] + sa * xa[o];
      Dout[o] = xres[o] + g * v;
    }
  }
}

// -------- MSFF: three dilated 3x3 depthwise convs + GELU gate --------
__global__ void dwconv_gate_kernel(const float* __restrict__ hT,
                                   const float* __restrict__ k1,
                                   const float* __restrict__ k2,
                                   const float* __restrict__ k3,
                                   float* __restrict__ gT) {
  int idx = blockIdx.x * blockDim.x + threadIdx.x;
  int pos = idx & (L_SEQ - 1);
  int c = (idx >> 12) % HID;
  int b = idx / (HID * L_SEQ);
  int yy = pos >> 6, xx = pos & 63;
  const float* P1 = hT + ((size_t)b * 3 * HID + c) * L_SEQ;
  const float* P2 = hT + ((size_t)b * 3 * HID + HID + c) * L_SEQ;
  const float* P3 = hT + ((size_t)b * 3 * HID + 2 * HID + c) * L_SEQ;
  const float* w1p = k1 + c * 9;
  const float* w2p = k2 + c * 9;
  const float* w3p = k3 + c * 9;
  float a1 = 0.f, a2 = 0.f, a3 = 0.f;
#pragma unroll
  for (int dy = -1; dy <= 1; ++dy) {
#pragma unroll
    for (int dx = -1; dx <= 1; ++dx) {
      int wi = (dy + 1) * 3 + (dx + 1);
      int y1 = yy + dy,     x1 = xx + dx;
      int y2 = yy + 2 * dy, x2 = xx + 2 * dx;
      int y3 = yy + 3 * dy, x3 = xx + 3 * dx;
      if ((unsigned)y1 < 64u && (unsigned)x1 < 64u) a1 += w1p[wi] * P1[y1 * 64 + x1];
      if ((unsigned)y2 < 64u && (unsigned)x2 < 64u) a2 += w2p[wi] * P2[y2 * 64 + x2];
      if ((unsigned)y3 < 64u && (unsigned)x3 < 64u) a3 += w3p[wi] * P3[y3 * 64 + x3];
    }
  }
  float ge = 0.5f * a1 * (1.f + erff(a1 * 0.70710678118654752f));  // exact GELU
  gT[((size_t)b * HID + c) * L_SEQ + pos] = ge * a2 * a3;
}

extern "C" void kernel_launch(void* const* d_in, const int* in_sizes, int n_in,
                              void* d_out, int out_size, void* d_ws, size_t ws_size,
                              hipStream_t stream) {
  (void)in_sizes; (void)n_in; (void)out_size; (void)ws_size;
  const float* x      = (const float*)d_in[0];
  const float* gamma1 = (const float*)d_in[1];
  const float* gamma2 = (const float*)d_in[2];
  const float* ln1w   = (const float*)d_in[3];
  const float* ln1b   = (const float*)d_in[4];
  const float* mnw    = (const float*)d_in[5];
  const float* mnb    = (const float*)d_in[6];
  const float* f_Win  = (const float*)d_in[7];
  const float* f_cw   = (const float*)d_in[8];
  const float* f_cb   = (const float*)d_in[9];
  const float* f_Wx   = (const float*)d_in[10];
  const float* f_Wdt  = (const float*)d_in[11];
  const float* f_bdt  = (const float*)d_in[12];
  const float* f_Alog = (const float*)d_in[13];
  const float* f_D    = (const float*)d_in[14];
  const float* b_Win  = (const float*)d_in[15];
  const float* b_cw   = (const float*)d_in[16];
  const float* b_cb   = (const float*)d_in[17];
  const float* b_Wx   = (const float*)d_in[18];
  const float* b_Wdt  = (const float*)d_in[19];
  const float* b_bdt  = (const float*)d_in[20];
  const float* b_Alog = (const float*)d_in[21];
  const float* b_D    = (const float*)d_in[22];
  const float* Wout   = (const float*)d_in[23];
  const float* m_win  = (const float*)d_in[24];
  const float* dw1    = (const float*)d_in[25];
  const float* dw2    = (const float*)d_in[26];
  const float* dw3    = (const float*)d_in[27];
  const float* m_wout = (const float*)d_in[28];

  // workspace layout (floats); hT aliases xzTf(+xzTb), gT aliases xsTf
  float* w = (float*)d_ws;
  const size_t SZ_PL = (size_t)NBATCH * DIM * L_SEQ;      // 786432
  const size_t SZ_DI = (size_t)NBATCH * DI * L_SEQ;       // 1572864
  const size_t SZ_XZ = (size_t)NBATCH * 2 * DI * L_SEQ;   // 3145728
  const size_t SZ_XD = (size_t)NBATCH * L_SEQ * 38;       // 311296
  float* x1   = w; w += SZ_PL;
  float* x2   = w; w += SZ_PL;
  float* yTf  = w; w += SZ_DI;
  float* yTb  = w; w += SZ_DI;
  float* xnT  = w; w += SZ_PL;   // reused as xmT
  float* xzTf = w; w += SZ_XZ;   // reused (with xzTb) as hT [B][576][L]
  float* xzTb = w; w += SZ_XZ;
  float* xsTf = w; w += SZ_DI;   // reused as gT
  float* xsTb = w; w += SZ_DI;
  float* xdf  = w; w += SZ_XD;
  float* xdb  = w; w += SZ_XD;
  float* hT = xzTf;
  float* gT = xsTf;

  // 1) LN chain: x1 = ln_cf(x), xn = ln(x1)
  ln_kernel<<<1024, 256, 0, stream>>>(x, ln1w, ln1b, mnw, mnb, x1, xnT);
  // 2) xz = xn @ Win.T (both directions, planar out [B][384][L])
  gemm_wact_kernel<<<768, 128, 0, stream>>>(f_Win, xnT, xzTf, 384, 96);
  gemm_wact_kernel<<<768, 128, 0, stream>>>(b_Win, xnT, xzTb, 384, 96);
  // 3) depthwise conv1d + silu (backward = anti-causal, reversal-free)
  conv1d_silu_kernel<<<6144, 256, 0, stream>>>(xzTf, f_cw, f_cb, xsTf, 0);
  conv1d_silu_kernel<<<6144, 256, 0, stream>>>(xzTb, b_cw, b_cb, xsTb, 1);
  // 4) x_dbl = xs @ Wx.T
  gemm_xdbl_kernel<<<128, 128, 0, stream>>>(xsTf, f_Wx, xdf);
  gemm_xdbl_kernel<<<128, 128, 0, stream>>>(xsTb, b_Wx, xdb);
  // 5) selective scan, both directions concurrently
  scan_kernel<<<48, 256, 0, stream>>>(xsTf, xdf, xzTf, f_Wdt, f_bdt, f_Alog, f_D,
                                      xsTb, xdb, xzTb, b_Wdt, b_bdt, b_Alog, b_D,
                                      yTf, yTb);
  // 6) x2 = x + gamma1*((yf+yb) @ Wout.T + x1)
  gemm_fuse_kernel<<<192, 128, 0, stream>>>(Wout, yTf, yTb, x, x1, gamma1, x2, 192);
  // 7) xm = ln_cf(x2)
  ln_kernel<<<1024, 256, 0, stream>>>(x2, ln1w, ln1b, nullptr, nullptr, xnT, nullptr);
  // 8) h = msff_win @ xm  (576 planes)
  gemm_wact_kernel<<<1152, 128, 0, stream>>>(m_win, xnT, hT, 576, 96);
  // 9) g = gelu(dw1(h0)) * dw2(h1) * dw3(h2)
  dwconv_gate_kernel<<<6144, 256, 0, stream>>>(hT, dw1, dw2, dw3, gT);
  // 10) out = x2 + gamma2 * (msff_wout @ g)
  gemm_fuse_kernel<<<192, 128, 0, stream>>>(m_wout, gT, nullptr, x2, nullptr, gamma2,
                                            (float*)d_out, 192);
}